// Digitcaps_10934986736332
// MI455X (gfx1250) — compile-verified
//
#include <hip/hip_runtime.h>
#include <math.h>

typedef __attribute__((ext_vector_type(2))) float v2f;
typedef __attribute__((ext_vector_type(8))) float v8f;

constexpr int B      = 256;   // batch
constexpr int ICAPS  = 3200;  // input capsules
constexpr int INDIM  = 8;     // n
constexpr int J      = 8;     // output capsules
constexpr int M      = 16;    // output dim
constexpr int BT     = 16;    // batch tile per block (WMMA M)
constexpr int IC     = 64;    // input capsules per block
constexpr int NIB    = ICAPS / IC;   // 50 chunks
constexpr int WAVES  = 8;
constexpr int TPB    = WAVES * 32;   // 256 threads (wave32)
constexpr int SSZ    = B * J * M;    // 32768 floats per s / v buffer
constexpr int UTS    = 18;           // padded row stride for u_hat transpose scratch

// ---------------------------------------------------------------------------
// Fused routing pass.
//  u_hat tile (16b x 16m) = X(16x8) @ W[i,j](8x16)  -> 2x v_wmma_f32_16x16x4_f32
//  agreement a[b] = diag( V(16x16) @ u_hat^T )      -> 4x v_wmma_f32_16x16x4_f32
//    (u_hat^T fragments gathered via a stride-18 LDS scratchpad; DS ops from
//     one wave are processed in order, so only compiler fences are needed)
//  softmax over j done by 16 owner lanes, c broadcast back through LDS.
//  s[b,j,m] = sum_i c*u_hat accumulated in regs -> LDS ds_add -> global atomics.
// vprev == nullptr  =>  iteration 0 (uniform c = 1/8, no agreement needed).
// ---------------------------------------------------------------------------
__global__ __launch_bounds__(TPB) void routing_pass(
    const float* __restrict__ x,      // [B][ICAPS][8]
    const float* __restrict__ W,      // [ICAPS][J][8][16]
    const float* __restrict__ vprev,  // [B][J][16] accumulated v, or nullptr
    float* __restrict__ s_out)        // [B][J][16], atomically accumulated
{
    __shared__ float xs[IC * BT * INDIM];      // 32 KB: x tile, [ii][bb][n]
    __shared__ float vs[BT * J * M];           //  8 KB: v tile, [bb][j][m]
    __shared__ float s_lds[J * BT * M];        //  8 KB: block s partials
    __shared__ float a_lds[WAVES * BT * J];    //  4 KB: logits -> c, per wave
    __shared__ float ut_lds[WAVES * BT * UTS]; //  9 KB: u_hat transpose pad-18

    const int t  = threadIdx.x;
    const int b0 = blockIdx.x * BT;
    const int i0 = blockIdx.y * IC;

    // ---- stage x tile (coalesced float4) ----
    {
        const float4* xg  = reinterpret_cast<const float4*>(x);
        float4*       xs4 = reinterpret_cast<float4*>(xs);
#pragma unroll
        for (int k = 0; k < (BT * IC * 2) / TPB; ++k) {
            const int q  = t + k * TPB;
            const int bb = q >> 7;
            const int r  = q & 127;
            const int ii = r >> 1;
            const int n4 = r & 1;
            xs4[(ii * BT + bb) * 2 + n4] =
                xg[((size_t)(b0 + bb) * ICAPS + (i0 + ii)) * 2 + n4];
        }
    }
    const bool uniform = (vprev == nullptr);
    if (!uniform) {
        const float4* vg  = reinterpret_cast<const float4*>(vprev) + b0 * (J * M / 4);
        float4*       vs4 = reinterpret_cast<float4*>(vs);
        for (int q = t; q < BT * J * M / 4; q += TPB) vs4[q] = vg[q];
    }
    for (int q = t; q < J * BT * M; q += TPB) s_lds[q] = 0.0f;
    __syncthreads();

    const int wave = t >> 5;
    const int lane = t & 31;
    const int half = lane >> 4;   // selects WMMA K rows / M+8 half
    const int lm   = lane & 15;   // A-row / B,C column within half

    float* up = &ut_lds[wave * (BT * UTS)];

    v8f sacc[J];
#pragma unroll
    for (int j = 0; j < J; ++j) sacc[j] = (v8f){0,0,0,0,0,0,0,0};

    for (int il = wave; il < IC; il += WAVES) {
        const int i = i0 + il;

        // A fragments (16x4 f32): A[b][k] = x[b][i][k]
        const float* xrow = &xs[(il * BT + lm) * INDIM];
        v2f af0, af1;
        af0.x = xrow[half * 2 + 0]; af0.y = xrow[half * 2 + 1];
        af1.x = xrow[half * 2 + 4]; af1.y = xrow[half * 2 + 5];

        // ---- u_hat tiles for all 8 classes ----
        v8f acc[J];
        const float* wbase = W + (size_t)i * (J * INDIM * M);
#pragma unroll
        for (int j = 0; j < J; ++j) {
            const float* wj = wbase + j * (INDIM * M);
            const int kr = half * 2;
            v2f bf0, bf1;   // B[k][m] = W[i][j][k][m]
            bf0.x = wj[(kr + 0) * M + lm]; bf0.y = wj[(kr + 1) * M + lm];
            bf1.x = wj[(kr + 4) * M + lm]; bf1.y = wj[(kr + 5) * M + lm];
            v8f c = {0,0,0,0,0,0,0,0};
            c = __builtin_amdgcn_wmma_f32_16x16x4_f32(false, af0, false, bf0,
                                                      (short)0, c, false, false);
            c = __builtin_amdgcn_wmma_f32_16x16x4_f32(false, af1, false, bf1,
                                                      (short)0, c, false, false);
            acc[j] = c;   // (r, lane) -> u_hat[b0 + r + 8*half][i][j][lm]
        }

        if (uniform) {
            // iteration 0: c = softmax(0) = 1/8
#pragma unroll
            for (int j = 0; j < J; ++j)
#pragma unroll
                for (int r = 0; r < 8; ++r) sacc[j][r] += 0.125f * acc[j][r];
        } else {
            // ---- agreement logits a[b] = diag(V . u_hat^T) via WMMA ----
            const bool holder = ((lm >> 3) == half);  // lanes owning diagonal
#pragma unroll
            for (int j = 0; j < J; ++j) {
                // bounce u_hat C-tile through padded LDS to gather U^T frags
#pragma unroll
                for (int r = 0; r < 8; ++r)
                    up[(r + half * 8) * UTS + lm] = acc[j][r];
                asm volatile("" ::: "memory");  // same-wave DS is in-order; fence compiler
                v8f d = {0,0,0,0,0,0,0,0};
#pragma unroll
                for (int kq = 0; kq < 4; ++kq) {
                    const int kb = kq * 4 + half * 2;
                    // A = V:   A[b=lm][k=m]   from vs (contiguous m -> b64)
                    v2f avf = *(const v2f*)&vs[(lm * J + j) * M + kb];
                    // B = U^T: B[k=m][n=lm] = U[lm][m]  (stride-18 rows, conflict-free)
                    v2f buf = *(const v2f*)&up[lm * UTS + kb];
                    d = __builtin_amdgcn_wmma_f32_16x16x4_f32(false, avf, false, buf,
                                                              (short)0, d, false, false);
                }
                // diagonal D[b][b] lives on holder lanes at register r = lm&7
                const int rsel = lm & 7;
                float dv = d[0];
#pragma unroll
                for (int r = 1; r < 8; ++r) dv = (rsel == r) ? d[r] : dv;
                if (holder) a_lds[(wave * BT + lm) * J + j] = dv;
                asm volatile("" ::: "memory");
            }

            // ---- softmax over j: one owner lane per batch row ----
            asm volatile("s_wait_dscnt 0x0" ::: "memory");
            if (lane < BT) {
                const int bb = lane;
                float av[J];
                float mx = -3.0e38f;
#pragma unroll
                for (int j = 0; j < J; ++j) {
                    av[j] = a_lds[(wave * BT + bb) * J + j];
                    mx = fmaxf(mx, av[j]);
                }
                float se = 0.0f;
#pragma unroll
                for (int j = 0; j < J; ++j) { av[j] = __expf(av[j] - mx); se += av[j]; }
                const float inv = 1.0f / se;
                v2f* cw = (v2f*)&a_lds[(wave * BT + bb) * J];
#pragma unroll
                for (int q = 0; q < 4; ++q) {
                    v2f c2; c2.x = av[2 * q] * inv; c2.y = av[2 * q + 1] * inv;
                    cw[q] = c2;
                }
            }
            asm volatile("s_wait_dscnt 0x0" ::: "memory");

            // ---- weight u_hat by c into s accumulators (packed b64 c reads) ----
#pragma unroll
            for (int r = 0; r < 8; ++r) {
                const int bb = r + half * 8;
                const v2f* cp = (const v2f*)&a_lds[(wave * BT + bb) * J];
#pragma unroll
                for (int q = 0; q < 4; ++q) {
                    v2f c2 = cp[q];
                    sacc[2 * q][r]     += c2.x * acc[2 * q][r];
                    sacc[2 * q + 1][r] += c2.y * acc[2 * q + 1][r];
                }
            }
        }
    }

    // ---- block reduction of s partials: registers -> LDS -> global atomics ----
#pragma unroll
    for (int j = 0; j < J; ++j)
#pragma unroll
        for (int r = 0; r < 8; ++r) {
            const int bb = r + half * 8;
            atomicAdd(&s_lds[(j * BT + bb) * M + lm], sacc[j][r]);
        }
    __syncthreads();
    for (int q = t; q < J * BT * M; q += TPB) {
        const int j  = q / (BT * M);
        const int bb = (q / M) & (BT - 1);
        const int m  = q & (M - 1);
        atomicAdd(&s_out[((size_t)(b0 + bb) * J + j) * M + m], s_lds[q]);
    }
}

// v = squash(s) (+ optional vadd for the accumulated-v trick); one thread per (b,j)
__global__ void squash_k(const float* __restrict__ s, const float* __restrict__ vadd,
                         float* __restrict__ vout)
{
    const int bj = blockIdx.x * blockDim.x + threadIdx.x;
    if (bj >= B * J) return;
    const float* sp = s + bj * M;
    float sq = 0.0f;
#pragma unroll
    for (int m = 0; m < M; ++m) { const float e = sp[m]; sq += e * e; }
    const float scale = sqrtf(sq) / (1.0f + sq);  // == sq/((1+sq)*sqrt(sq))
#pragma unroll
    for (int m = 0; m < M; ++m) {
        float r = sp[m] * scale;
        if (vadd) r += vadd[bj * M + m];
        vout[bj * M + m] = r;
    }
}

__global__ void zero_ws(float* p, int n)
{
    const int t = blockIdx.x * blockDim.x + threadIdx.x;
    if (t < n) p[t] = 0.0f;
}

extern "C" void kernel_launch(void* const* d_in, const int* in_sizes, int n_in,
                              void* d_out, int out_size, void* d_ws, size_t ws_size,
                              hipStream_t stream)
{
    (void)in_sizes; (void)n_in; (void)out_size; (void)ws_size;
    const float* x = (const float*)d_in[0];   // [256][3200][8]
    const float* W = (const float*)d_in[1];   // [3200][8][8][16]
    float* out = (float*)d_out;               // [256][8][16]
    float* ws  = (float*)d_ws;

    float* s0  = ws;             // s, iteration 0
    float* s1  = ws + SSZ;       // s, iteration 1
    float* s2  = ws + 2 * SSZ;   // s, iteration 2
    float* v0  = ws + 3 * SSZ;   // v0
    float* v01 = ws + 4 * SSZ;   // v0 + v1 (accumulated agreement direction)

    zero_ws<<<(3 * SSZ + TPB - 1) / TPB, TPB, 0, stream>>>(ws, 3 * SSZ);

    const dim3 grid(B / BT, NIB);  // 16 batch-tiles x 50 capsule chunks = 800 blocks
    // iter 0: c uniform
    routing_pass<<<grid, TPB, 0, stream>>>(x, W, nullptr, s0);
    squash_k<<<(B * J + TPB - 1) / TPB, TPB, 0, stream>>>(s0, nullptr, v0);
    // iter 1: logits = u_hat . v0
    routing_pass<<<grid, TPB, 0, stream>>>(x, W, v0, s1);
    squash_k<<<(B * J + TPB - 1) / TPB, TPB, 0, stream>>>(s1, v0, v01);  // v01 = v0 + v1
    // iter 2 (final): logits = u_hat . (v0 + v1)
    routing_pass<<<grid, TPB, 0, stream>>>(x, W, v01, s2);
    squash_k<<<(B * J + TPB - 1) / TPB, TPB, 0, stream>>>(s2, nullptr, out);
}